// MutualMultiAttentionHead_90357521973758
// MI455X (gfx1250) — compile-verified
//
#include <hip/hip_runtime.h>
#include <hip/hip_bf16.h>
#include <math.h>

// Problem constants (from reference)
#define NH 4
#define NF 64
#define FE 16
#define NN 4096
#define NE 65536
#define CW (NH * NF)   // 256, node_lin columns

typedef __attribute__((ext_vector_type(2))) float v2f;
typedef __attribute__((ext_vector_type(8))) float v8f;

// ---------------------------------------------------------------------------
// Kernel 1: node projection GEMM  C[4096,256] = A[4096,64] @ Wn[64,256] + bn
// fp32 WMMA (V_WMMA_F32_16X16X4_F32): exact precision match to reference.
// One wave per 16x16 tile; K=64 -> 16 wmma steps of K=4.
// ---------------------------------------------------------------------------
__global__ __launch_bounds__(256)
void node_proj_wmma(const float* __restrict__ A, const float* __restrict__ W,
                    const float* __restrict__ bn, float* __restrict__ C) {
  const int wave = threadIdx.x >> 5;
  const int lane = threadIdx.x & 31;
  const int tile = blockIdx.x * 8 + wave;   // 4096 tiles = 256 mtiles x 16 ntiles
  const int mt = tile >> 4;
  const int nt = tile & 15;
  const int m0 = mt * 16, n0 = nt * 16;
  const int lm = lane & 15;
  const int hi = lane >> 4;                  // half-wave select
  const int n  = n0 + lm;

  v8f acc = {};
#if __has_builtin(__builtin_amdgcn_wmma_f32_16x16x4_f32)
  const int kb = hi * 2;                     // K offset within each 4-step
  const float* Arow = A + (m0 + lm) * NF;    // A row for this lane's M
  #pragma unroll
  for (int s = 0; s < 16; ++s) {
    const int k = s * 4 + kb;
    v2f a; a.x = Arow[k];           a.y = Arow[k + 1];            // A[m, k..k+1]
    v2f b; b.x = W[k * CW + n];     b.y = W[(k + 1) * CW + n];    // B[k..k+1, n]
    acc = __builtin_amdgcn_wmma_f32_16x16x4_f32(
        /*neg_a=*/false, a, /*neg_b=*/false, b,
        /*c_mod=*/(short)0, acc, /*reuse_a=*/false, /*reuse_b=*/false);
  }
#else
  // VALU fallback (keeps compiling if builtin absent; same output layout)
  #pragma unroll
  for (int j = 0; j < 8; ++j) {
    const int m = m0 + hi * 8 + j;
    float t = 0.f;
    for (int k = 0; k < NF; ++k) t += A[m * NF + k] * W[k * CW + n];
    acc[j] = t;
  }
#endif
  const float bias = bn[n];
  #pragma unroll
  for (int j = 0; j < 8; ++j) {
    const int m = m0 + hi * 8 + j;           // C/D layout: VGPR j -> row m0+hi*8+j
    C[m * CW + n] = acc[j] + bias;
  }
}

// ---------------------------------------------------------------------------
// Kernel 2: edge logits. One thread per edge_lin row r (65536 rows).
// edge_proj[h, e] = edge_lin[h*16384 + e/4, e%4]; scaled by 1/sqrt(16)=0.25.
// Written directly in attn layout z[h*NE + e].
// ---------------------------------------------------------------------------
__global__ __launch_bounds__(256)
void edge_logits(const float* __restrict__ EIn, const float* __restrict__ We,
                 const float* __restrict__ be, float* __restrict__ z) {
  const int r = blockIdx.x * blockDim.x + threadIdx.x;
  if (r >= NE) return;
  const float* x = EIn + r * FE;
  float acc[NH] = {be[0], be[1], be[2], be[3]};
  #pragma unroll
  for (int j = 0; j < FE; ++j) {
    const float xv = x[j];
    #pragma unroll
    for (int c = 0; c < NH; ++c) acc[c] += xv * We[j * NH + c];
  }
  const int h = r >> 14;          // r / 16384
  const int q = r & 16383;        // r % 16384
  float4* out = (float4*)(z + h * NE + q * 4);
  *out = make_float4(acc[0] * 0.25f, acc[1] * 0.25f, acc[2] * 0.25f, acc[3] * 0.25f);
}

// ---------------------------------------------------------------------------
// Kernel 3: per-head softmax over all 65536 edges. One 1024-thread block/head.
// In-place: z (logits) -> attn. 3 phases: max, sum, normalize. wave32 shuffles.
// ---------------------------------------------------------------------------
__global__ __launch_bounds__(1024)
void head_softmax(float* __restrict__ z) {
  const int h = blockIdx.x;
  float* zh = z + h * NE;
  const int t = threadIdx.x;
  __shared__ float red[32];

  float m = -1e30f;
  for (int i = t; i < NE; i += 1024) m = fmaxf(m, zh[i]);
  #pragma unroll
  for (int off = 16; off > 0; off >>= 1) m = fmaxf(m, __shfl_down(m, off, 32));
  if ((t & 31) == 0) red[t >> 5] = m;
  __syncthreads();
  if (t < 32) {
    float v = red[t];
    #pragma unroll
    for (int off = 16; off > 0; off >>= 1) v = fmaxf(v, __shfl_down(v, off, 32));
    if (t == 0) red[0] = v;
  }
  __syncthreads();
  const float gmax = red[0];
  __syncthreads();

  float s = 0.f;
  for (int i = t; i < NE; i += 1024) s += __expf(zh[i] - gmax);
  #pragma unroll
  for (int off = 16; off > 0; off >>= 1) s += __shfl_down(s, off, 32);
  if ((t & 31) == 0) red[t >> 5] = s;
  __syncthreads();
  if (t < 32) {
    float v = red[t];
    #pragma unroll
    for (int off = 16; off > 0; off >>= 1) v += __shfl_down(v, off, 32);
    if (t == 0) red[0] = v;
  }
  __syncthreads();
  const float inv = 1.f / red[0];
  for (int i = t; i < NE; i += 1024) zh[i] = __expf(zh[i] - gmax) * inv;
}

// ---------------------------------------------------------------------------
// Kernel 4a: zero the output (harness poisons d_out).
// ---------------------------------------------------------------------------
__global__ __launch_bounds__(256)
void zero_out(float* __restrict__ out, int n) {
  const int i = blockIdx.x * blockDim.x + threadIdx.x;
  if (i < n) out[i] = 0.f;
}

// ---------------------------------------------------------------------------
// Kernel 4b: sparse message passing + head-mean via f32 global atomics.
// One wave per edge; each lane handles 2 of the 64 features.
// node_proj[h,d,f] = node_lin[h*1024 + d/4, (d%4)*64 + f]  (torch .view quirk)
// ---------------------------------------------------------------------------
__global__ __launch_bounds__(256)
void scatter_msg(const float* __restrict__ node_lin, const float* __restrict__ attn,
                 const int* __restrict__ src, const int* __restrict__ dst,
                 float* __restrict__ out) {
  const int wave = threadIdx.x >> 5;
  const int lane = threadIdx.x & 31;
  const int e = blockIdx.x * 8 + wave;
  if (e >= NE) return;
  const int s = src[e];
  const int d = dst[e];
  const int f0 = lane * 2;
  float acc0 = 0.f, acc1 = 0.f;
  #pragma unroll
  for (int h = 0; h < NH; ++h) {
    const float a = attn[h * NE + e] * 0.25f;                 // includes 1/H mean
    const float* row = node_lin + (h * 1024 + (d >> 2)) * CW + (d & 3) * NF;
    acc0 += a * row[f0];
    acc1 += a * row[f0 + 1];
  }
  atomicAdd(&out[s * NF + f0], acc0);
  atomicAdd(&out[s * NF + f0 + 1], acc1);
}

// ---------------------------------------------------------------------------
extern "C" void kernel_launch(void* const* d_in, const int* in_sizes, int n_in,
                              void* d_out, int out_size, void* d_ws, size_t ws_size,
                              hipStream_t stream) {
  const float* node_inputs = (const float*)d_in[0];   // [4096,64]
  const float* edge_inputs = (const float*)d_in[1];   // [65536,16]
  const int*   src         = (const int*)  d_in[2];   // [65536]
  const int*   dst         = (const int*)  d_in[3];   // [65536]
  const float* Wn          = (const float*)d_in[4];   // [64,256]
  const float* bn          = (const float*)d_in[5];   // [256]
  const float* We          = (const float*)d_in[6];   // [16,4]
  const float* be          = (const float*)d_in[7];   // [4]
  float* out = (float*)d_out;                         // [4096,64]

  // Workspace layout: node_lin [4096*256] | attn/z [4*65536]  (5 MB total)
  float* node_lin = (float*)d_ws;
  float* z        = node_lin + NN * CW;

  // Node projection GEMM (WMMA): 4096 tiles, 8 waves/block -> 512 blocks
  node_proj_wmma<<<512, 256, 0, stream>>>(node_inputs, Wn, bn, node_lin);

  // Edge logits: 65536 rows / 256
  edge_logits<<<NE / 256, 256, 0, stream>>>(edge_inputs, We, be, z);

  // Per-head softmax (in-place z -> attn)
  head_softmax<<<NH, 1024, 0, stream>>>(z);

  // Zero output then scatter-accumulate messages
  zero_out<<<(NN * NF + 255) / 256, 256, 0, stream>>>(out, NN * NF);
  scatter_msg<<<NE / 8, 256, 0, stream>>>(node_lin, z, src, dst, out);
}